// RealCamera_69853348102833
// MI455X (gfx1250) — compile-verified
//
#include <hip/hip_runtime.h>

// CDNA5 / gfx1250: wave32, WMMA 16x16x4 f32.
typedef float v2f __attribute__((ext_vector_type(2)));
typedef float v8f __attribute__((ext_vector_type(8)));

static constexpr double factd(int n) {
  double f = 1.0;
  for (int i = 2; i <= n; ++i) f *= (double)i;
  return f;
}

// One lane per pose. Block = 256 threads = 8 waves. Per wave: 32 poses,
// 8 x V_WMMA_F32_16X16X4_F32 handle the (W @ proj^T) 4x4 GEMMs (4 poses each),
// with W staged through padded LDS into the WMMA B layout.
__global__ __launch_bounds__(256) void pose_update_kernel(
    const float* __restrict__ g_w,
    const float* __restrict__ g_v,
    const float* __restrict__ g_S,
    float* __restrict__ g_out,
    int n)
{
  __shared__ float sW[256 * 17];  // 17-float stride: bank-conflict padding

  const int tid  = threadIdx.x;
  const int lane = tid & 31;
  const int wave = tid >> 5;
  const int gid  = blockIdx.x * 256 + tid;
  const int i    = (gid < n) ? gid : (n - 1);  // clamp keeps EXEC all-ones for WMMA

  // ---- loads: twist (3+3 dwords) + start pose (3 x b128) ----
  const float w0 = g_w[3 * i + 0], w1 = g_w[3 * i + 1], w2 = g_w[3 * i + 2];
  const float vv0 = g_v[3 * i + 0], vv1 = g_v[3 * i + 1], vv2 = g_v[3 * i + 2];
  const float4* S4 = (const float4*)g_S;
  const float4 s0 = S4[(size_t)i * 4 + 0];
  const float4 s1 = S4[(size_t)i * 4 + 1];
  const float4 s2 = S4[(size_t)i * 4 + 2];
  // row 3 of an SE(3) pose is [0,0,0,1]; never needed.

  // ---- Taylor A=sin/x, B=(1-cos)/x^2, C=(x-sin)/x^3 : Horner in theta^2 ----
  const float x2 = w0 * w0 + w1 * w1 + w2 * w2;
  float A = 0.f, B = 0.f, C = 0.f;
#pragma unroll
  for (int k = 10; k >= 0; --k) {
    const float sgn = (k & 1) ? -1.f : 1.f;
    A = fmaf(A, x2, sgn * (float)(1.0 / factd(2 * k + 1)));
    B = fmaf(B, x2, sgn * (float)(1.0 / factd(2 * k + 2)));
    C = fmaf(C, x2, sgn * (float)(1.0 / factd(2 * k + 3)));
  }

  // ---- R = I + A*wx + B*wx2 ; V = I + B*wx + C*wx2 ; wx2 = w w^T - x2*I ----
  const float wvec[3] = {w0, w1, w2};
  const float wx[3][3] = {{0.f, -w2, w1}, {w2, 0.f, -w0}, {-w1, w0, 0.f}};
  float Rm[3][3], Vm[3][3];
#pragma unroll
  for (int r = 0; r < 3; ++r)
#pragma unroll
    for (int c = 0; c < 3; ++c) {
      const float o  = wvec[r] * wvec[c] - ((r == c) ? x2 : 0.f);
      const float id = (r == c) ? 1.f : 0.f;
      Rm[r][c] = id + A * wx[r][c] + B * o;
      Vm[r][c] = id + B * wx[r][c] + C * o;
    }
  float tt[3];
#pragma unroll
  for (int r = 0; r < 3; ++r)
    tt[r] = Vm[r][0] * vv0 + Vm[r][1] * vv1 + Vm[r][2] * vv2;

  // ---- pose_w2c = S @ inv(deltaT):  Rp = Rs*R^T,  tp = ts - Rp*t ----
  const float Rs[3][3] = {{s0.x, s0.y, s0.z}, {s1.x, s1.y, s1.z}, {s2.x, s2.y, s2.z}};
  const float ts[3] = {s0.w, s1.w, s2.w};
  float Rp[3][3], tp[3];
#pragma unroll
  for (int r = 0; r < 3; ++r)
#pragma unroll
    for (int c = 0; c < 3; ++c)
      Rp[r][c] = Rs[r][0] * Rm[c][0] + Rs[r][1] * Rm[c][1] + Rs[r][2] * Rm[c][2];
#pragma unroll
  for (int r = 0; r < 3; ++r)
    tp[r] = ts[r] - (Rp[r][0] * tt[0] + Rp[r][1] * tt[1] + Rp[r][2] * tt[2]);

  // ---- W = pose_w2c^T ----
  float W[4][4];
#pragma unroll
  for (int r = 0; r < 3; ++r) {
#pragma unroll
    for (int c = 0; c < 3; ++c) W[r][c] = Rp[c][r];
    W[r][3] = 0.f;
  }
  W[3][0] = tp[0]; W[3][1] = tp[1]; W[3][2] = tp[2]; W[3][3] = 1.f;

  // ---- output 1: world_view_transform (4 x b128 per pose) ----
  float4* outW = (float4*)g_out;
#pragma unroll
  for (int r = 0; r < 4; ++r) {
    float4 row;
    row.x = W[r][0]; row.y = W[r][1]; row.z = W[r][2]; row.w = W[r][3];
    outW[(size_t)i * 4 + r] = row;
  }

  // ---- output 3: camera_center = -Rp^T * tp ----
  float* outCC = g_out + (size_t)n * 32;
#pragma unroll
  for (int c = 0; c < 3; ++c)
    outCC[(size_t)i * 3 + c] =
        -(Rp[0][c] * tp[0] + Rp[1][c] * tp[1] + Rp[2][c] * tp[2]);

  // ---- stage W to LDS for WMMA B-operand gather ----
  const int sbase = tid * 17;
#pragma unroll
  for (int r = 0; r < 4; ++r)
#pragma unroll
    for (int c = 0; c < 4; ++c) sW[sbase + r * 4 + c] = W[r][c];
  __syncthreads();

  // ---- output 2: full_proj^T = proj @ W^T via V_WMMA_F32_16X16X4_F32 ----
  // proj rows: {ax,0,0,0},{0,ax,0,0},{0,0,cz,dz},{0,0,1,0}  (fovx==fovy)
  const float ax = 1.0f / tanf(0.4f);
  const float cz = 10.0f / 9.9f;
  const float dz = -1.0f / 9.9f;

  // A (16x4) = proj stacked 4x. ISA A-layout: lanes 0-15 hold K=0,1 (VGPR0,1)
  // for M=lane; lanes 16-31 hold K=2,3 for M=lane-16.
  const int r4 = lane & 3;
  v2f aop;
  if (lane < 16) {
    aop[0] = (r4 == 0) ? ax : 0.f;   // proj[r][0]
    aop[1] = (r4 == 1) ? ax : 0.f;   // proj[r][1]
  } else {
    aop[0] = (r4 == 2) ? cz : ((r4 == 3) ? 1.f : 0.f);  // proj[r][2]
    aop[1] = (r4 == 2) ? dz : 0.f;                      // proj[r][3]
  }

  // B (4x16) = [W0^T W1^T W2^T W3^T]. B-layout (mirrors C/D column striping):
  // lane n (0-15): VGPR0=B[0][n], VGPR1=B[1][n]; lanes 16-31: K=2,3.
  // B[k][4j+c] = W_j[c][k]  -> 2 dword LDS reads per lane per chunk.
  const int nloc = lane & 15;
  const int jj   = nloc >> 2;
  const int cc   = nloc & 3;
  const int k0   = (lane < 16) ? 0 : 2;
  float* outF = g_out + (size_t)n * 16;

#pragma unroll
  for (int q = 0; q < 8; ++q) {
    const int pib = wave * 32 + q * 4 + jj;          // pose within block
    const int lb  = pib * 17 + cc * 4 + k0;
    v2f bop;
    bop[0] = sW[lb];
    bop[1] = sW[lb + 1];
    v8f acc = {0.f, 0.f, 0.f, 0.f, 0.f, 0.f, 0.f, 0.f};
    // D[4i+r][4j+c] = (proj @ W_j^T)[r][c] = full_j[c][r]  (same for every i)
    acc = __builtin_amdgcn_wmma_f32_16x16x4_f32(
        /*neg_a=*/false, aop, /*neg_b=*/false, bop,
        /*c_mod=*/(short)0, acc, /*reuse_a=*/false, /*reuse_b=*/false);
    // Lane N=4j+c (N<16) holds full_j[c][0..3] in acc[0..3] -> one b128 row store.
    if (lane < 16) {
      int gp = blockIdx.x * 256 + wave * 32 + q * 4 + (lane >> 2);
      if (gp > n - 1) gp = n - 1;
      float4 row;
      row.x = acc[0]; row.y = acc[1]; row.z = acc[2]; row.w = acc[3];
      ((float4*)(outF + (size_t)gp * 16))[lane & 3] = row;
    }
  }
}

extern "C" void kernel_launch(void* const* d_in, const int* in_sizes, int n_in,
                              void* d_out, int out_size, void* d_ws, size_t ws_size,
                              hipStream_t stream) {
  (void)n_in; (void)out_size; (void)d_ws; (void)ws_size;
  const float* w = (const float*)d_in[0];
  const float* v = (const float*)d_in[1];
  const float* S = (const float*)d_in[2];
  const int n = in_sizes[0] / 3;
  const int blocks = (n + 255) / 256;
  pose_update_kernel<<<blocks, 256, 0, stream>>>(w, v, S, (float*)d_out, n);
}